// label_aware_dual_MSEloss_29540785062440
// MI455X (gfx1250) — compile-verified
//
#include <hip/hip_runtime.h>
#include <hip/hip_bf16.h>
#include <math.h>

typedef float v2f __attribute__((ext_vector_type(2)));
typedef float v8f __attribute__((ext_vector_type(8)));

#define B_N   8192
#define NT    256
#define NQ    7
#define REF_Y 0.5f
#define ALPHA 0.5f
#define EPS   1e-8f

__global__ __launch_bounds__(NT) void label_aware_dual_mse_loss_kernel(
    const float* __restrict__ x,   // input  [B,1]
    const float* __restrict__ t,   // target [B,1]
    float* __restrict__ out)       // scalar loss
{
    __shared__ float part[NQ][NT];   // per-thread partials, 7 KB LDS

    const int tid = threadIdx.x;

    // ---------------- Phase 1: per-thread partial sums (strided, coalesced) ----
    float s0 = 0.f, s1 = 0.f, s2 = 0.f, s3 = 0.f, s4 = 0.f, s5 = 0.f, s6 = 0.f;

    #pragma unroll 4
    for (int i = tid; i < B_N; i += NT) {
        const float ti = t[i];
        const float xi = x[i];

        const float ad    = fabsf(REF_Y - ti);
        const float denom = fabsf(REF_Y) + fabsf(ti) + EPS;
        float w = ad / denom;
        w = fminf(fmaxf(w, 0.1f), 2.0f);        // clip(.., 0.1, 2.0)

        const bool  valid = !(ti != ti);         // !isnan(target)
        const float e  = xi - ti;                // yh - y
        const float e2 = e * e;
        const float wv = valid ? w : 0.0f;

        s0 += wv;            // sum(wv)           (MSE denominator)
        s1 += wv * e2;       // sum(se*wv)        (MSE numerator; (t-x)^2 == e^2)
        s2 += w;             // sum(w)
        s3 += e;             // sum(e)
        s4 += e2;            // sum(e^2)
        s5 += w * e;         // sum(w*e)
        s6 += w * e2;        // sum(w*e^2)
    }

    part[0][tid] = s0; part[1][tid] = s1; part[2][tid] = s2; part[3][tid] = s3;
    part[4][tid] = s4; part[5][tid] = s5; part[6][tid] = s6;
    __syncthreads();

    // ---------------- Phase 2: wave 0 reduces 256 -> 1 per quantity via WMMA ---
    // A = ones(16x4). D = A x B + C  =>  every row of D holds the column sums of
    // B, so the grand total of D row 0 equals the total of all 64 B elements,
    // independent of the exact (k,n) slot each value occupies.  Four chained
    // WMMAs fold all 256 partials into one accumulator.
    if (tid < 32) {                       // exactly wave 0: EXEC all-ones
        v2f a; a.x = 1.0f; a.y = 1.0f;    // ones A-operand (16x4 f32, 2 VGPRs)

        float tot[NQ];
        #pragma unroll
        for (int q = 0; q < NQ; ++q) {
            v8f d = {};                   // C = 0
            #pragma unroll
            for (int c = 0; c < 4; ++c) { // 4 chunks x 64 values = 256
                v2f b;
                b.x = part[q][c * 64 + tid];
                b.y = part[q][c * 64 + 32 + tid];
                d = __builtin_amdgcn_wmma_f32_16x16x4_f32(
                        /*neg_a=*/false, a, /*neg_b=*/false, b,
                        /*c_mod=*/(short)0, d,
                        /*reuse_a=*/false, /*reuse_b=*/false);
            }
            // D row 0 lives in d[0]: lane n (n<16) = column-sum n (lanes 16..31
            // mirror row 8, identical values).  Sum the 16-lane group.
            float v = d[0];
            v += __shfl_xor(v, 1, 32);
            v += __shfl_xor(v, 2, 32);
            v += __shfl_xor(v, 4, 32);
            v += __shfl_xor(v, 8, 32);
            tot[q] = v;                   // lane 0 holds sum of lanes 0..15
        }

        // ---------------- Phase 3: scalar epilogue on lane 0 -------------------
        if (tid == 0) {
            const float S_wv   = tot[0];
            const float S_sewv = tot[1];
            const float S_w    = tot[2];
            const float S_e    = tot[3];
            const float S_e2   = tot[4];
            const float S_we   = tot[5];
            const float S_we2  = tot[6];

            const float mse = S_sewv / S_wv;

            const float Bf  = (float)B_N;
            // closed form of the upper-triangle pairwise sums
            const float num = 0.5f * (Bf * S_we2 + S_e2 * S_w - 2.0f * S_we * S_e);
            const float den = 0.5f * (Bf - 1.0f) * S_w;
            float contrastive = num / den;

            const bool do_scale = (contrastive > EPS) && (mse > EPS) &&
                                  (contrastive < mse);
            const float scale = do_scale ? (mse / contrastive) : 1.0f;
            contrastive *= scale;

            out[0] = ALPHA * mse + (1.0f - ALPHA) * contrastive;
        }
    }
}

extern "C" void kernel_launch(void* const* d_in, const int* in_sizes, int n_in,
                              void* d_out, int out_size, void* d_ws, size_t ws_size,
                              hipStream_t stream)
{
    (void)in_sizes; (void)n_in; (void)out_size; (void)d_ws; (void)ws_size;
    const float* input  = (const float*)d_in[0];
    const float* target = (const float*)d_in[1];
    float* out = (float*)d_out;

    // 64 KB of input total: single workgroup, single kernel — launch-latency
    // bound, so avoid multi-kernel / atomic designs entirely.
    label_aware_dual_mse_loss_kernel<<<1, NT, 0, stream>>>(input, target, out);
}